// GroupQueryAttentionDecodeLayer_9311489098083
// MI455X (gfx1250) — compile-verified
//
#include <hip/hip_runtime.h>
#include <math.h>

// Problem constants (from reference)
#define B_ 16
#define H_ 32
#define G_ 8
#define S_ 8192
#define D_ 128
#define HQ 4            // query heads per KV group
#define SPLITS 32       // flash-decode splits over S
#define WPB 2           // waves per block
#define CHUNK (S_ / SPLITS)   // 256 keys per wave
#define TILES (CHUNK / 16)    // 16 key-tiles of 16 keys
#define LD 132                // LDS row stride (floats), D+4 vs bank conflicts
#define GD (G_ * D_)          // global row stride between consecutive s

typedef float v2f __attribute__((ext_vector_type(2)));
typedef float v4f __attribute__((ext_vector_type(4)));
typedef float v8f __attribute__((ext_vector_type(8)));
typedef int   v4i __attribute__((ext_vector_type(4)));

__device__ __forceinline__ v8f wmma_f32_16x16x4(v2f a, v2f b, v8f c) {
  // D = A(16x4 f32) * B(4x16 f32) + C(16x16 f32)
  return __builtin_amdgcn_wmma_f32_16x16x4_f32(false, a, false, b, (short)0, c,
                                               false, false);
}

// ---- CDNA5 async global->LDS staging (ASYNCcnt), with sync fallback --------
#if defined(__has_builtin)
#if __has_builtin(__builtin_amdgcn_global_load_async_to_lds_b128)
#define USE_ASYNC_LDS 1
#endif
#endif
#ifndef USE_ASYNC_LDS
#define USE_ASYNC_LDS 0
#endif

// Issue one 16-key x 128-d tile (16 rows, one b128 per lane per row).
__device__ __forceinline__ void tile_load_issue(const float* __restrict__ g,
                                                float* lds, int lane) {
#pragma unroll
  for (int r = 0; r < 16; ++r) {
    const float* src = g + (size_t)r * GD + lane * 4;
    float* dst = lds + r * LD + lane * 4;
#if USE_ASYNC_LDS
    __builtin_amdgcn_global_load_async_to_lds_b128((v4i*)src, (v4i*)dst, 0, 0);
#else
    *(v4f*)dst = __builtin_nontemporal_load((const v4f*)src);
#endif
  }
}

template <int N>
__device__ __forceinline__ void wait_async() {
#if USE_ASYNC_LDS
#if __has_builtin(__builtin_amdgcn_s_wait_asynccnt)
  __builtin_amdgcn_s_wait_asynccnt(N);
#else
  asm volatile("s_wait_asynccnt %0" ::"i"(N) : "memory");
#endif
#endif
}
// ---------------------------------------------------------------------------

__global__ __launch_bounds__(WPB * 32) void gqa_partial(
    const float* __restrict__ q, const float* __restrict__ kc,
    const float* __restrict__ vc, float* __restrict__ opart,
    float* __restrict__ mpart, float* __restrict__ lpart) {
  __shared__ float qs[HQ * D_];            // scaled Q for this (b,g)
  __shared__ float kbuf[WPB][16 * LD];     // per-wave K tile
  __shared__ float vbuf[WPB][16 * LD];     // per-wave V tile
  __shared__ float pbuf[WPB][16 * HQ];     // per-wave P^T (key x head)

  const int b = blockIdx.z, g = blockIdx.y;
  const int tid = threadIdx.x;
  const int wave = tid >> 5, lane = tid & 31;
  const int hlo = lane & 15;            // row/col index within 16-lane half
  const int hi2 = (lane >> 4) << 1;     // K-offset: 0 (low half) / 2 (high half)

  const float scale = 0.08838834764831845f;  // 1/sqrt(128), folded into Q
  for (int i = tid; i < HQ * D_; i += WPB * 32)
    qs[i] = q[((size_t)b * H_ + g * HQ + (i >> 7)) * D_ + (i & 127)] * scale;
  __syncthreads();

  // Q A-fragments resident in registers: 32 slices of K=4 along D.
  // A 16x4 f32 layout: lanes 0-15 hold (M=lane, K=0/1) in v.x/v.y,
  //                    lanes 16-31 hold (M=lane-16, K=2/3). Rows >= HQ zero.
  v2f qa[32];
#pragma unroll
  for (int ks = 0; ks < 32; ++ks) {
    const int d0 = 4 * ks + hi2;
    float x = 0.f, y = 0.f;
    if (hlo < HQ) { x = qs[hlo * D_ + d0]; y = qs[hlo * D_ + d0 + 1]; }
    qa[ks] = (v2f){x, y};
  }

  const int split = blockIdx.x * WPB + wave;
  const size_t s0 = (size_t)split * CHUNK;
  const float* kbg = kc + ((size_t)b * S_) * GD + (size_t)g * D_;
  const float* vbg = vc + ((size_t)b * S_) * GD + (size_t)g * D_;
  float* kb = kbuf[wave];
  float* vb = vbuf[wave];
  float* pw = pbuf[wave];

  v8f acc[8];  // O accumulator: comp r lanes0-15 = (head r, d = 16t+lane)
#pragma unroll
  for (int t = 0; t < 8; ++t) acc[t] = (v8f){0, 0, 0, 0, 0, 0, 0, 0};
  float m_run[HQ] = {-INFINITY, -INFINITY, -INFINITY, -INFINITY};
  float l_run[HQ] = {0.f, 0.f, 0.f, 0.f};

  // ---- software pipeline: prefetch K(0)
  tile_load_issue(kbg + s0 * GD, kb, lane);

#pragma unroll 1
  for (int tile = 0; tile < TILES; ++tile) {
    const size_t srow = s0 + (size_t)tile * 16;

    wait_async<0>();                       // K(tile) landed in kb
    tile_load_issue(vbg + srow * GD, vb, lane);  // V(tile) flies during QK

    // ---- scores: C = Q * K^T  (M=16 heads [4 real], N=16 keys), 32 WMMAs
    v8f c = (v8f){0, 0, 0, 0, 0, 0, 0, 0};
#pragma unroll
    for (int ks = 0; ks < 32; ++ks) {
      const int d0 = 4 * ks + hi2;
      // B 4x16 layout mirrors A: v.x/v.y = K-rows (0,1) low half, (2,3) high
      v2f bf = (v2f){kb[hlo * LD + d0], kb[hlo * LD + d0 + 1]};
      c = wmma_f32_16x16x4(qa[ks], bf, c);
    }

    // ---- online softmax per real head r (keys live across lanes 0-15)
    float corr[HQ];
#pragma unroll
    for (int r = 0; r < HQ; ++r) {
      float mr = c[r];
#pragma unroll
      for (int mask = 8; mask >= 1; mask >>= 1)
        mr = fmaxf(mr, __shfl_xor(mr, mask, 32));
      const float mt = __shfl(mr, 0, 32);
      const float nm = fmaxf(m_run[r], mt);
      corr[r] = __expf(m_run[r] - nm);
      m_run[r] = nm;
    }
    v4f pv;
#pragma unroll
    for (int r = 0; r < HQ; ++r) {
      const float pr = __expf(c[r] - m_run[r]);
      pv[r] = pr;
      float sr = pr;
#pragma unroll
      for (int mask = 8; mask >= 1; mask >>= 1) sr += __shfl_xor(sr, mask, 32);
      const float ts = __shfl(sr, 0, 32);
      l_run[r] = l_run[r] * corr[r] + ts;
    }
    // rescale O (only real-head components are ever nonzero)
#pragma unroll
    for (int t = 0; t < 8; ++t)
#pragma unroll
      for (int r = 0; r < HQ; ++r) acc[t][r] *= corr[r];

    // P^T (key x head) to LDS for the PV A-fragments (lane = key)
    if (lane < 16) *(v4f*)(pw + lane * 4) = pv;

    // Prefetch K(tile+1); all QK reads of kb were consumed by the WMMAs above.
    // Async completion is in-order, so waiting to <=16 outstanding guarantees
    // the 16 V-rows (issued first) have landed while K(tile+1) keeps flying.
    if (tile + 1 < TILES) {
      tile_load_issue(kbg + (srow + 16) * GD, kb, lane);
      wait_async<16>();                    // V(tile) landed in vb
    } else {
      wait_async<0>();                     // last tile: V(tile) landed
    }

    // ---- O += P * V  (M=16 heads, N=16 d-cols x 8 tiles, K=4 keys), 32 WMMAs
#pragma unroll
    for (int k0 = 0; k0 < 16; k0 += 4) {
      const int kk = k0 + hi2;
      float ax = 0.f, ay = 0.f;
      if (hlo < HQ) { ax = pw[kk * HQ + hlo]; ay = pw[(kk + 1) * HQ + hlo]; }
      v2f a = (v2f){ax, ay};
#pragma unroll
      for (int t = 0; t < 8; ++t) {
        const int col = t * 16 + hlo;
        v2f bf = (v2f){vb[kk * LD + col], vb[(kk + 1) * LD + col]};
        acc[t] = wmma_f32_16x16x4(a, bf, acc[t]);
      }
    }
  }

  // ---- write unnormalized per-split partials + (m, l)
  const size_t base = ((size_t)(b * G_ + g) * SPLITS + split) * HQ;
  if (lane < 16) {
#pragma unroll
    for (int t = 0; t < 8; ++t)
#pragma unroll
      for (int r = 0; r < HQ; ++r)
        opart[(base + r) * D_ + t * 16 + lane] = acc[t][r];
  }
  if (lane == 0) {
#pragma unroll
    for (int r = 0; r < HQ; ++r) {
      mpart[base + r] = m_run[r];
      lpart[base + r] = l_run[r];
    }
  }
}

// Combine SPLITS partials: one block per (b,h), one thread per d.
__global__ __launch_bounds__(D_) void gqa_reduce(
    const float* __restrict__ opart, const float* __restrict__ mpart,
    const float* __restrict__ lpart, float* __restrict__ out) {
  const int bh = blockIdx.x;
  const int d = threadIdx.x;
  const int b = bh / H_, h = bh % H_;
  const int g = h / HQ, hi = h % HQ;
  const size_t pbase = (size_t)(b * G_ + g) * SPLITS * HQ + hi;

  float M = -INFINITY;
#pragma unroll 1
  for (int sp = 0; sp < SPLITS; ++sp)
    M = fmaxf(M, mpart[pbase + (size_t)sp * HQ]);

  float L = 0.f, accd = 0.f;
#pragma unroll 1
  for (int sp = 0; sp < SPLITS; ++sp) {
    const float w = __expf(mpart[pbase + (size_t)sp * HQ] - M);
    L += w * lpart[pbase + (size_t)sp * HQ];
    accd += w * opart[(pbase + (size_t)sp * HQ) * D_ + d];
  }
  out[(size_t)bh * D_ + d] = accd / L;
}

extern "C" void kernel_launch(void* const* d_in, const int* in_sizes, int n_in,
                              void* d_out, int out_size, void* d_ws,
                              size_t ws_size, hipStream_t stream) {
  const float* q = (const float*)d_in[0];
  const float* k = (const float*)d_in[1];
  const float* v = (const float*)d_in[2];
  float* out = (float*)d_out;

  // Workspace: O partials [B,G,SPLITS,HQ,D] + m,l [B,G,SPLITS,HQ]  (~8.2 MB)
  float* ws = (float*)d_ws;
  float* opart = ws;
  float* mpart = ws + (size_t)B_ * G_ * SPLITS * HQ * D_;
  float* lpart = mpart + (size_t)B_ * G_ * SPLITS * HQ;

  dim3 grid1(SPLITS / WPB, G_, B_);  // 16 x 8 x 16 = 2048 blocks, 2 waves each
  gqa_partial<<<grid1, WPB * 32, 0, stream>>>(q, k, v, opart, mpart, lpart);
  gqa_reduce<<<B_ * H_, D_, 0, stream>>>(opart, mpart, lpart, out);
}